// MIMOLayer_46462956208720
// MI455X (gfx1250) — compile-verified
//
#include <hip/hip_runtime.h>
#include <math.h>

// ---- problem constants (match reference) ----
#define BSZ  16
#define SEQL 1024
#define HD   256
#define PD   256
#define BL   (BSZ * SEQL)   // 16384 rows

typedef __attribute__((ext_vector_type(2))) float v2f;
typedef __attribute__((ext_vector_type(4))) float v4f;
typedef __attribute__((ext_vector_type(8))) float v8f;

// Fragment layout for the WMMA B operand (4xK rows x 16 cols per step).
// Assumed mirror of the documented A layout:
//   VGPR0: lanes 0-15 -> K=k+0, lanes 16-31 -> K=k+2
//   VGPR1: lanes 0-15 -> K=k+1, lanes 16-31 -> K=k+3
// Packed flat so a wave reads one contiguous float2 per lane per k-step.
__device__ __forceinline__ int frag_off(int k, int n, int ntiles) {
    int kt = k >> 2, kr = k & 3;
    int nt = n >> 4, nr = n & 15;
    int lane = nr + ((kr & 2) << 3);       // +16 for K=k+2/k+3
    return ((kt * ntiles + nt) * 32 + lane) * 2 + (kr & 1);
}

// ---------------- prep: per-mode ZOH discretization ----------------
__global__ __launch_bounds__(256)
void prep_modes(const float* __restrict__ Lre, const float* __restrict__ Lim,
                const float* __restrict__ logstep,
                float* __restrict__ lamb_re, float* __restrict__ lamb_im,
                float* __restrict__ coef_re, float* __restrict__ coef_im) {
    int p = threadIdx.x;
    float dt = expf(logstep[p]);
    float lr = Lre[p], li = Lim[p];
    float er = expf(lr * dt);
    float s, c;
    sincosf(li * dt, &s, &c);
    float lbr = er * c, lbi = er * s;          // Lambda_bar = exp(Lambda*dt)
    float nr = lbr - 1.0f, ni = lbi;           // Lambda_bar - 1
    float den = lr * lr + li * li;             // |Lambda|^2
    lamb_re[p] = lbr;
    lamb_im[p] = lbi;
    coef_re[p] = (nr * lr + ni * li) / den;    // (Lambda_bar-1)/Lambda
    coef_im[p] = (ni * lr - nr * li) / den;
}

// ---------------- pack B_bar into WMMA fragments ----------------
// Logical weight Wb[k=h][n], n<PD -> B_bar_re[p=n][h], else B_bar_im[p=n-PD][h]
__global__ __launch_bounds__(256)
void pack_wb(const float* __restrict__ Bp,
             const float* __restrict__ coef_re, const float* __restrict__ coef_im,
             float* __restrict__ Wb) {
    int idx = blockIdx.x * 256 + threadIdx.x;      // 256*512 elements
    int k = idx >> 9;            // h
    int n = idx & 511;
    int p = n & 255;
    float cr = coef_re[p], ci = coef_im[p];
    float br = Bp[(p * HD + k) * 2 + 0];
    float bi = Bp[(p * HD + k) * 2 + 1];
    float val = (n < PD) ? (cr * br - ci * bi)     // Re(coef * B_tilde)
                         : (cr * bi + ci * br);    // Im(coef * B_tilde)
    Wb[frag_off(k, n, 512 / 16)] = val;
}

// Logical weight Wc[k][h], k<PD -> 2*C_re[h][p=k], else -2*C_im[h][p=k-PD]
__global__ __launch_bounds__(256)
void pack_wc(const float* __restrict__ Cp, float* __restrict__ Wc) {
    int idx = blockIdx.x * 256 + threadIdx.x;      // 512*256 elements
    int k = idx >> 8;
    int h = idx & 255;
    int p = k & 255;
    float cre = Cp[(h * PD + p) * 2 + 0];
    float cim = Cp[(h * PD + p) * 2 + 1];
    float val = (k < PD) ? (2.0f * cre) : (-2.0f * cim);
    Wc[frag_off(k, h, 256 / 16)] = val;
}

// ---------------- fp32 WMMA GEMM: Out[M x N] = A[M x K] @ Wfrag ----------------
// Block = 4 waves; each wave computes one 16x16 tile; block covers 16 rows x 64 cols.
template <int K, int N, bool FEED>
__global__ __launch_bounds__(128)
void gemm_wmma(const float* __restrict__ A, const float* __restrict__ Wf,
               float* __restrict__ Out,
               const float* __restrict__ uin, const float* __restrict__ Dv) {
    __shared__ float As[16 * (K + 4)];             // +4 float pad -> no LDS bank conflicts
    const int tid = threadIdx.x;
    const int m0 = blockIdx.x * 16;

    // Stage 16 x K A-strip into LDS with coalesced float4 loads.
    constexpr int K4 = K / 4;
    const float* Arow = A + (size_t)m0 * K;
    for (int i = tid; i < 16 * K4; i += 128) {
        int r = i / K4, c = (i % K4) * 4;
        *(v4f*)(&As[r * (K + 4) + c]) = *(const v4f*)(Arow + (size_t)r * K + c);
    }
    __syncthreads();

    const int lane  = tid & 31;
    const int wave  = tid >> 5;
    const int ntile = blockIdx.y * 4 + wave;
    const int m     = lane & 15;
    const int ksel  = (lane >> 4) << 1;            // 0 or 2 (upper K pair)
    const float* wptr = Wf + (size_t)ntile * 64 + lane * 2;

    v8f acc = {};
#pragma unroll 8
    for (int k = 0; k < K; k += 4) {
        v2f a = *(const v2f*)(&As[m * (K + 4) + k + ksel]);
        v2f b = *(const v2f*)(wptr + (size_t)(k >> 2) * (N / 16) * 64);
        // D = A(16x4,f32) * B(4x16,f32) + C ; 8 args: neg_a,A,neg_b,B,c_mod,C,reuse_a,reuse_b
        acc = __builtin_amdgcn_wmma_f32_16x16x4_f32(false, a, false, b,
                                                    (short)0, acc, false, false);
    }

    // C/D layout: VGPR j -> M=j (lanes 0-15) / M=8+j (lanes 16-31), N=lane&15.
    const int row0 = m0 + ((lane >> 4) << 3);
    const int col  = ntile * 16 + (lane & 15);
#pragma unroll
    for (int j = 0; j < 8; ++j) {
        int row = row0 + j;
        float v = acc[j];
        if (FEED)   // fused feedthrough: + D[h] * u[row][h]
            v = fmaf(Dv[col], uin[(size_t)row * HD + col], v);
        Out[(size_t)row * N + col] = v;
    }
}

// ---------------- complex diagonal recurrence (in-place over Bu) ----------------
// x_t = Lambda_bar * x_{t-1} + Bu_t ; one thread per (batch, mode).
__global__ __launch_bounds__(32)
void scan_kernel(float* __restrict__ Bu,
                 const float* __restrict__ lamr, const float* __restrict__ lami) {
    int g = blockIdx.x * 32 + threadIdx.x;         // 0 .. 4095
    int b = g >> 8, p = g & 255;
    float lr = lamr[p], li = lami[p];
    float xr = 0.0f, xi = 0.0f;
    float* base = Bu + (size_t)b * SEQL * 512 + p; // re at +0, im at +256
    for (int l = 0; l < SEQL; ++l) {
        size_t off = (size_t)l * 512;
        float br = base[off];
        float bi = base[off + 256];
        float nxr = fmaf(lr, xr, fmaf(-li, xi, br));
        float nxi = fmaf(lr, xi, fmaf( li, xr, bi));
        xr = nxr; xi = nxi;
        base[off]       = xr;
        base[off + 256] = xi;
    }
}

// ---------------- launcher ----------------
extern "C" void kernel_launch(void* const* d_in, const int* in_sizes, int n_in,
                              void* d_out, int out_size, void* d_ws, size_t ws_size,
                              hipStream_t stream) {
    const float* u       = (const float*)d_in[0];  // (16,1024,256)
    const float* Lre     = (const float*)d_in[1];  // (256,)
    const float* Lim     = (const float*)d_in[2];  // (256,)
    const float* Bp      = (const float*)d_in[3];  // (256,256,2)
    const float* Cp      = (const float*)d_in[4];  // (256,256,2)
    const float* Dv      = (const float*)d_in[5];  // (256,)
    const float* logstep = (const float*)d_in[6];  // (256,1)
    float* out = (float*)d_out;                    // (16,1024,256)

    float* ws      = (float*)d_ws;
    float* lamb_re = ws;                 // 256
    float* lamb_im = ws + 256;           // 256
    float* coef_re = ws + 512;           // 256
    float* coef_im = ws + 768;           // 256
    float* Wb      = ws + 1024;          // 256*512 packed fragments
    float* Wc      = Wb + 256 * 512;     // 512*256 packed fragments
    float* Bu      = Wc + 512 * 256;     // BL*512 (re ‖ im), scan runs in-place

    prep_modes<<<1, 256, 0, stream>>>(Lre, Lim, logstep,
                                      lamb_re, lamb_im, coef_re, coef_im);
    pack_wb<<<512, 256, 0, stream>>>(Bp, coef_re, coef_im, Wb);
    pack_wc<<<512, 256, 0, stream>>>(Cp, Wc);

    // Bu[bl, 0:256]=Re, [256:512]=Im : (16384 x 256) @ (256 x 512)
    gemm_wmma<256, 512, false><<<dim3(BL / 16, 512 / 64), 128, 0, stream>>>(
        u, Wb, Bu, nullptr, nullptr);

    scan_kernel<<<(BSZ * PD) / 32, 32, 0, stream>>>(Bu, lamb_re, lamb_im);

    // y = xs(16384 x 512) @ Wc(512 x 256) + D*u
    gemm_wmma<512, 256, true><<<dim3(BL / 16, 256 / 64), 128, 0, stream>>>(
        Bu, Wc, out, u, Dv);
}